// MultitaskReadout_12859132084760
// MI455X (gfx1250) — compile-verified
//
#include <hip/hip_runtime.h>

typedef float v2f __attribute__((ext_vector_type(2)));
typedef float v4f __attribute__((ext_vector_type(4)));
typedef float v8f __attribute__((ext_vector_type(8)));

#define B_    16
#define N_    4096
#define DIM_  1024
#define T_    8
#define E_    64
#define NTOK  (B_ * N_)        // 65536 tokens
#define KSTEPS (DIM_ / 4)      // 256 WMMA K-steps of 4
#define WIN   1024             // tokens per binning workgroup
#define NBINWG (NTOK / WIN)    // 64
#define MAXG  4608             // max 16-token groups (64 WGs * 71 worst case)
#define GPW   8                // groups (waves) per GEMM block

// d_ws layout (bytes)
#define CNT_OFF   0
#define WSW_OFF   256
#define WSW_BYTES (T_ * KSTEPS * 32 * 8 * 4)   // 2 MB swizzled W
#define GH_OFF    (WSW_OFF + WSW_BYTES)
#define GT_OFF    (GH_OFF + MAXG * 4)

// ---------------------------------------------------------------------------
// Kernel 1: swizzle W[T][DIM][E] into WMMA-B lane order.
// For (head h, k-step s, lane L): 8 floats = for tile c=0..3 the B-operand
// pair ( W[4s+koff][c*16+ln], W[4s+koff+1][c*16+ln] ), koff = (L>=16)?2:0.
// Also resets the global group counter.
// ---------------------------------------------------------------------------
__global__ __launch_bounds__(256) void reformat_w(const float* __restrict__ W,
                                                  float* __restrict__ wsw,
                                                  int* __restrict__ counter) {
    if (blockIdx.x == 0 && threadIdx.x == 0) *counter = 0;
    int o = blockIdx.x * 256 + threadIdx.x;     // 0 .. T_*KSTEPS*32*8 - 1
    int f    = o & 7;
    int lane = (o >> 3) & 31;
    int s    = (o >> 8) & (KSTEPS - 1);
    int h    = o >> 16;
    int c    = f >> 1;
    int pair = f & 1;
    int ln   = lane & 15;
    int koff = (lane >> 4) << 1;
    int k = 4 * s + koff + pair;
    int e = c * 16 + ln;
    wsw[o] = W[(size_t)h * DIM_ * E_ + (size_t)k * E_ + e];
}

// ---------------------------------------------------------------------------
// Kernel 2: bin WIN tokens by head; per-head lists padded to multiples of 16
// (pad slot = -1); append whole 16-token single-head groups to global list.
// ---------------------------------------------------------------------------
__global__ __launch_bounds__(256) void bin_tokens(const int* __restrict__ idx,
                                                  int* __restrict__ counter,
                                                  int* __restrict__ gheads,
                                                  int* __restrict__ gtoks) {
    __shared__ int cnt[T_];
    __shared__ int cnt2[T_];
    __shared__ int offs[T_ + 1];
    __shared__ int flat[WIN + T_ * 15];
    __shared__ int baseg, ptot;

    int tid = threadIdx.x;
    for (int i = tid; i < WIN + T_ * 15; i += 256) flat[i] = -1;
    if (tid < T_) { cnt[tid] = 0; cnt2[tid] = 0; }
    __syncthreads();

    int tbase = blockIdx.x * WIN;
    int h[4];
    #pragma unroll
    for (int j = 0; j < 4; ++j) {
        int t = tbase + tid * 4 + j;
        h[j] = idx[t] & 7;
        atomicAdd(&cnt[h[j]], 1);
    }
    __syncthreads();

    if (tid == 0) {
        int run = 0;
        for (int q = 0; q < T_; ++q) {
            offs[q] = run;
            run += (cnt[q] + 15) & ~15;
        }
        offs[T_] = run;
        ptot = run;
        baseg = atomicAdd(counter, run >> 4);
    }
    __syncthreads();

    #pragma unroll
    for (int j = 0; j < 4; ++j) {
        int t = tbase + tid * 4 + j;
        int pos = atomicAdd(&cnt2[h[j]], 1);
        flat[offs[h[j]] + pos] = t;
    }
    __syncthreads();

    int pt = ptot, bg = baseg;
    for (int i = tid; i < pt; i += 256) gtoks[(bg << 4) + i] = flat[i];
    int ngl = pt >> 4;
    for (int gi = tid; gi < ngl; gi += 256) {
        int x = gi << 4;
        int hh = 0;
        #pragma unroll
        for (int q = 0; q < T_; ++q)
            if (offs[q] <= x && x < offs[q + 1]) hh = q;
        gheads[bg + gi] = hh;
    }
}

// ---------------------------------------------------------------------------
// Kernel 3: one wave per 16-token group: [16 x 1024] @ [1024 x 64] via
// v_wmma_f32_16x16x4_f32, 4 column tiles, 256 K-steps. Pad rows (tok = -1)
// compute garbage but are never stored, so EXEC stays all-1s for the WMMAs.
// ---------------------------------------------------------------------------
__global__ __launch_bounds__(256) void readout_gemm(
    const float* __restrict__ emb, const float* __restrict__ bias,
    const float* __restrict__ wsw, const int* __restrict__ counter,
    const int* __restrict__ gheads, const int* __restrict__ gtoks,
    float* __restrict__ out) {

    int wave = threadIdx.x >> 5;
    int lane = threadIdx.x & 31;
    int g = blockIdx.x * GPW + wave;
    int ng = *counter;
    if (g >= ng) return;

    int head = gheads[g];
    int ln   = lane & 15;
    int koff = (lane >> 4) << 1;                 // lanes 0-15: K+0,K+1 ; 16-31: K+2,K+3
    int tok  = gtoks[(g << 4) + ln];
    int tokA = tok < 0 ? 0 : tok;

    const float* aptr = emb + (size_t)tokA * DIM_ + koff;
    const float* bptr = wsw + ((size_t)head * KSTEPS * 32 + lane) * 8;

    v8f c0 = {}, c1 = {}, c2 = {}, c3 = {};

    #pragma unroll 4
    for (int s = 0; s < KSTEPS; ++s) {
        v2f a   = *(const v2f*)(aptr + 4 * s);
        v4f b01 = *(const v4f*)(bptr + (size_t)s * 256);
        v4f b23 = *(const v4f*)(bptr + (size_t)s * 256 + 4);
        v2f bt0 = __builtin_shufflevector(b01, b01, 0, 1);
        v2f bt1 = __builtin_shufflevector(b01, b01, 2, 3);
        v2f bt2 = __builtin_shufflevector(b23, b23, 0, 1);
        v2f bt3 = __builtin_shufflevector(b23, b23, 2, 3);
        c0 = __builtin_amdgcn_wmma_f32_16x16x4_f32(false, a, false, bt0, (short)0, c0, false, false);
        c1 = __builtin_amdgcn_wmma_f32_16x16x4_f32(false, a, false, bt1, (short)0, c1, false, false);
        c2 = __builtin_amdgcn_wmma_f32_16x16x4_f32(false, a, false, bt2, (short)0, c2, false, false);
        c3 = __builtin_amdgcn_wmma_f32_16x16x4_f32(false, a, false, bt3, (short)0, c3, false, false);
    }

    float bv0 = bias[head * E_ +  0 + ln];
    float bv1 = bias[head * E_ + 16 + ln];
    float bv2 = bias[head * E_ + 32 + ln];
    float bv3 = bias[head * E_ + 48 + ln];

    // D layout: VGPR r holds row M = r (lanes 0-15, N = lane) / M = r+8 (lanes 16-31, N = lane-16)
    #pragma unroll
    for (int r = 0; r < 8; ++r) {
        int m  = r + ((lane >> 4) << 3);
        int tm = gtoks[(g << 4) + m];
        if (tm >= 0) {
            float* op = out + (size_t)tm * E_ + ln;
            op[0]  = c0[r] + bv0;
            op[16] = c1[r] + bv1;
            op[32] = c2[r] + bv2;
            op[48] = c3[r] + bv3;
        }
    }
}

extern "C" void kernel_launch(void* const* d_in, const int* in_sizes, int n_in,
                              void* d_out, int out_size, void* d_ws, size_t ws_size,
                              hipStream_t stream) {
    const float* emb  = (const float*)d_in[0];   // [B, N, DIM] f32
    const int*   idx  = (const int*)d_in[1];     // [B, N] int
    const float* W    = (const float*)d_in[2];   // [T, DIM, E] f32
    const float* bias = (const float*)d_in[3];   // [T, E] f32
    float* out = (float*)d_out;                  // [B, N, E] f32

    char* ws = (char*)d_ws;
    int*   counter = (int*)(ws + CNT_OFF);
    float* wsw     = (float*)(ws + WSW_OFF);
    int*   gheads  = (int*)(ws + GH_OFF);
    int*   gtoks   = (int*)(ws + GT_OFF);

    hipLaunchKernelGGL(reformat_w, dim3((T_ * KSTEPS * 32 * 8) / 256), dim3(256), 0, stream,
                       W, wsw, counter);
    hipLaunchKernelGGL(bin_tokens, dim3(NBINWG), dim3(256), 0, stream,
                       idx, counter, gheads, gtoks);
    hipLaunchKernelGGL(readout_gemm, dim3(MAXG / GPW), dim3(256), 0, stream,
                       emb, bias, wsw, counter, gheads, gtoks, out);
}